// SAGPool_28424093565725
// MI455X (gfx1250) — compile-verified
//
#include <hip/hip_runtime.h>
#include <math.h>

#define G_   128
#define N_   1024
#define D_   256
#define NT_  (G_ * N_)        // 131072
#define E_   (NT_ * 32)       // 4194304
#define K_   (N_ / 2)         // 512
#define GK_  (G_ * K_)        // 65536

typedef float v2f __attribute__((ext_vector_type(2)));
typedef float v8f __attribute__((ext_vector_type(8)));

// ---------------- init: deg=1 (self loop), score=0 ----------------
__global__ void init_kernel(float* __restrict__ deg, float* __restrict__ score) {
    int i = blockIdx.x * blockDim.x + threadIdx.x;
    if (i < NT_) { deg[i] = 1.0f; score[i] = 0.0f; }
}

// ---------------- deg[row] += ew ----------------
__global__ void deg_kernel(const int* __restrict__ row, const float* __restrict__ ew,
                           float* __restrict__ deg) {
    int e = blockIdx.x * blockDim.x + threadIdx.x;
    if (e < E_) atomicAdd(&deg[row[e]], ew[e]);
}

// ---------------- deg -> dinv (deg >= 1 always, matches reference branch) ----
__global__ void dinv_kernel(float* __restrict__ deg) {
    int i = blockIdx.x * blockDim.x + threadIdx.x;
    if (i < NT_) {
        float d = deg[i];
        deg[i] = (d > 0.0f) ? rsqrtf(fmaxf(d, 1e-12f)) : 0.0f;
    }
}

// ---------------- h = x @ W via V_WMMA_F32_16X16X4_F32 ----------------
// One wave computes 16 rows. A = 16x4 tile of x (lane l, vgpr j -> K=2*(l>>4)+j),
// B = 4x16 with W broadcast across all 16 columns, C accumulates over K=256.
__global__ void gemv_wmma_kernel(const float* __restrict__ x,
                                 const float* __restrict__ W,
                                 float* __restrict__ h) {
    __shared__ __align__(16) float wsh[D_];
    for (int i = threadIdx.x; i < D_; i += blockDim.x) wsh[i] = W[i];
    __syncthreads();

    const int lane = threadIdx.x & 31;
    const int wave = threadIdx.x >> 5;
    const int base = blockIdx.x * 128 + wave * 16;   // 8 waves * 16 rows = 128 rows/block
    const int m    = lane & 15;
    const int koff = (lane >> 4) << 1;               // 0 or 2
    const float* xrow = x + (size_t)(base + m) * D_;

    v8f c = {};
#pragma unroll 4
    for (int k0 = 0; k0 < D_; k0 += 4) {
        v2f a = *(const v2f*)(xrow + k0 + koff);     // 8B-aligned (k0+koff even)
        v2f b = *(const v2f*)(&wsh[k0 + koff]);
        c = __builtin_amdgcn_wmma_f32_16x16x4_f32(
                /*neg_a=*/false, a, /*neg_b=*/false, b,
                /*c_mod=*/(short)0, c, /*reuse_a=*/false, /*reuse_b=*/false);
    }
    // All D columns are identical; c[r] holds row r (lanes 0-15) / row 8+r (lanes 16-31)
    if (lane == 0) {
#pragma unroll
        for (int r = 0; r < 8; ++r) h[base + r] = c[r];
    } else if (lane == 16) {
#pragma unroll
        for (int r = 0; r < 8; ++r) h[base + 8 + r] = c[r];
    }
}

// ---------------- score[row] += h[col]*dinv[row]*ew*dinv[col] ----------------
__global__ void edge_score_kernel(const int* __restrict__ row, const int* __restrict__ col,
                                  const float* __restrict__ ew, const float* __restrict__ h,
                                  const float* __restrict__ dinv, float* __restrict__ score) {
    int e = blockIdx.x * blockDim.x + threadIdx.x;
    if (e >= E_) return;
    int r = row[e], c = col[e];
    atomicAdd(&score[r], h[c] * dinv[r] * ew[e] * dinv[c]);
}

// ---------------- score += self-loop term + bias ----------------
__global__ void finalize_score_kernel(const float* __restrict__ h,
                                      const float* __restrict__ dinv,
                                      const float* __restrict__ b,
                                      float* __restrict__ score) {
    int i = blockIdx.x * blockDim.x + threadIdx.x;
    if (i < NT_) {
        float di = dinv[i];
        score[i] = score[i] + h[i] * di * di + b[0];
    }
}

// ---------------- per-graph top-K by ranking + prefix compaction ----------------
__global__ void __launch_bounds__(N_)
select_kernel(const float* __restrict__ score, int* __restrict__ new_id,
              int* __restrict__ sel, int* __restrict__ ngi_out) {
    __shared__ float s[N_];
    __shared__ int   p[N_];
    const int g = blockIdx.x;
    const int t = threadIdx.x;
    const int node = g * N_ + t;
    const float st = score[node];
    s[t] = st;
    __syncthreads();

    int rank = 0;
    for (int j = 0; j < N_; ++j) {                 // LDS broadcast reads
        float sj = s[j];
        rank += (sj > st) || (sj == st && j < t);  // jax top_k tiebreak: lower idx wins
    }
    const int msk = (rank < K_) ? 1 : 0;           // exactly K_ winners (ranks are a perm)

    p[t] = msk;
    __syncthreads();
    for (int off = 1; off < N_; off <<= 1) {       // Hillis-Steele inclusive scan
        int v = (t >= off) ? p[t - off] : 0;
        __syncthreads();
        p[t] += v;
        __syncthreads();
    }
    const int excl = p[t] - msk;                   // == sorted position within graph

    if (msk) {
        int nid = g * K_ + excl;
        new_id[node] = nid;
        sel[nid]     = node;
        ngi_out[nid] = g;
    } else {
        new_id[node] = -1;
    }
}

// ---------------- new_x[nid] = x[sel[nid]] * tanh(score[sel[nid]]) ----------------
__global__ void gather_kernel(const float* __restrict__ x, const float* __restrict__ score,
                              const int* __restrict__ sel, float* __restrict__ out_x) {
    __shared__ float tsh;
    __shared__ int   nodesh;
    const int nid = blockIdx.x;
    if (threadIdx.x == 0) {
        int node = sel[nid];
        nodesh = node;
        tsh = tanhf(score[node]);
    }
    __syncthreads();
    const int node = nodesh;
    out_x[(size_t)nid * D_ + threadIdx.x] = x[(size_t)node * D_ + threadIdx.x] * tsh;
}

// ---------------- re-index edges ----------------
__global__ void edge_out_kernel(const int* __restrict__ row, const int* __restrict__ col,
                                const float* __restrict__ ew, const int* __restrict__ new_id,
                                int* __restrict__ out_ei, float* __restrict__ out_ew) {
    int e = blockIdx.x * blockDim.x + threadIdx.x;
    if (e >= E_) return;
    int nr = new_id[row[e]];
    int nc = new_id[col[e]];
    bool valid = (nr >= 0) && (nc >= 0);
    out_ei[e]       = valid ? nr : 0;
    out_ei[E_ + e]  = valid ? nc : 0;
    out_ew[e]       = valid ? ew[e] : 0.0f;
}

extern "C" void kernel_launch(void* const* d_in, const int* in_sizes, int n_in,
                              void* d_out, int out_size, void* d_ws, size_t ws_size,
                              hipStream_t stream) {
    (void)in_sizes; (void)n_in; (void)out_size; (void)ws_size;

    const float* x   = (const float*)d_in[0];          // [NT_, D_]
    const int*   ei  = (const int*)d_in[1];            // [2, E_]: row=ei, col=ei+E_
    const float* ew  = (const float*)d_in[2];          // [E_]
    /* d_in[3]: node_graph_index — g is derivable, unused */
    const float* W   = (const float*)d_in[4];          // [D_]
    const float* b   = (const float*)d_in[5];          // [1]
    const int* row = ei;
    const int* col = ei + E_;

    // workspace layout
    float* deg    = (float*)d_ws;                      // NT_ floats (becomes dinv in-place)
    float* h      = deg + NT_;                         // NT_
    float* score  = h + NT_;                           // NT_
    int*   new_id = (int*)(score + NT_);               // NT_
    int*   sel    = new_id + NT_;                      // GK_ used

    // output layout (ints stored bitwise)
    float* out_x   = (float*)d_out;                    // GK_*D_
    int*   out_ei  = (int*)(out_x + (size_t)GK_ * D_); // 2*E_
    float* out_ew  = (float*)(out_ei + 2 * (size_t)E_);// E_
    int*   out_ngi = (int*)(out_ew + E_);              // GK_

    init_kernel          <<<NT_ / 256, 256, 0, stream>>>(deg, score);
    deg_kernel           <<<E_  / 256, 256, 0, stream>>>(row, ew, deg);
    dinv_kernel          <<<NT_ / 256, 256, 0, stream>>>(deg);
    gemv_wmma_kernel     <<<NT_ / 128, 256, 0, stream>>>(x, W, h);
    edge_score_kernel    <<<E_  / 256, 256, 0, stream>>>(row, col, ew, h, deg, score);
    finalize_score_kernel<<<NT_ / 256, 256, 0, stream>>>(h, deg, b, score);
    select_kernel        <<<G_,        N_,  0, stream>>>(score, new_id, sel, out_ngi);
    gather_kernel        <<<GK_,       D_,  0, stream>>>(x, score, sel, out_x);
    edge_out_kernel      <<<E_  / 256, 256, 0, stream>>>(row, col, ew, new_id, out_ei, out_ew);
}